// Detector_65747359367349
// MI455X (gfx1250) — compile-verified
//
#include <hip/hip_runtime.h>
#include <hip/hip_bf16.h>

typedef __attribute__((ext_vector_type(2)))  float    v2f;
typedef __attribute__((ext_vector_type(8)))  float    v8f;

#define NA   2000     // total anchors per image
#define NL0  1600     // level-0 anchors (40x40)
#define NC   20       // classes
#define NGTS 100      // gt slots
#define BTHREADS 256  // 8 wave32 per block

__device__ __forceinline__ float sigm(float x){ return 1.0f/(1.0f + __expf(-x)); }

// stable BCE-with-logits: log(1+e^x) - x*t
__device__ __forceinline__ float bce_logits(float x, float t){
  return fmaxf(x, 0.0f) - x*t + __logf(1.0f + __expf(-fabsf(x)));
}

// monotonic float<->uint mapping (order-preserving for all finite floats)
__device__ __forceinline__ unsigned f2mono(float f){
  unsigned u = __float_as_uint(f);
  return (u & 0x80000000u) ? ~u : (u | 0x80000000u);
}
__device__ __forceinline__ float mono2f(unsigned m){
  unsigned u = (m & 0x80000000u) ? (m & 0x7FFFFFFFu) : ~m;
  return __uint_as_float(u);
}

__device__ __forceinline__ unsigned long long shflxor64(unsigned long long v, int off){
  unsigned lo = (unsigned)v, hi = (unsigned)(v >> 32);
  lo = __shfl_xor(lo, off, 32);
  hi = __shfl_xor(hi, off, 32);
  return ((unsigned long long)hi << 32) | lo;
}

__device__ __forceinline__ void anchor_geom(int a, float& axc, float& ayc, float& st){
  int local, W;
  if (a < NL0){ local = a;       W = 40; st = 16.f; }
  else        { local = a - NL0; W = 20; st = 32.f; }
  axc = ((float)(local % W) + 0.5f) * st;
  ayc = ((float)(local / W) + 0.5f) * st;
}

__global__ __launch_bounds__(BTHREADS)
void yolox_loss_kernel(const float* __restrict__ l0, const float* __restrict__ l1,
                       const float* __restrict__ gboxes, const int* __restrict__ gcls,
                       const int* __restrict__ ngt, float* __restrict__ acc)
{
  // LDS-resident per-image state (~90KB; fits MI455X 320KB WGP LDS, 1 WG/WGP)
  __shared__ float sBX[NA], sBY[NA], sBW[NA], sBH[NA];   // decoded boxes (pixels)
  __shared__ float sObj[NA];                             // sigmoid(obj logit)
  __shared__ float sS[NA];                               // -sum_j log(1-p_j+eps)
  __shared__ float sGx[NGTS], sGy[NGTS], sGw[NGTS], sGh[NGTS];
  __shared__ int   sGc[NGTS], sGv[NGTS];
  __shared__ unsigned char sFg[NA];                      // fg candidate flag
  __shared__ int   sCnt[NA];                             // match count per anchor
  __shared__ int   sSel[NA];                             // matched gt (valid iff cnt==1)
  __shared__ unsigned long long sBest[NA];               // packed (cost,m) argmin per anchor
  __shared__ unsigned sMask[8][64];                      // per-wave selection bitmask
  __shared__ float sRed[4][BTHREADS];                    // reduction staging

  const int b    = blockIdx.x;
  const int tid  = threadIdx.x;
  const int lane = tid & 31;
  const int wave = tid >> 5;

  const float* base0 = l0 + (size_t)b * 25 * NL0;
  const float* base1 = l1 + (size_t)b * 25 * 400;

  // ---------------- Phase 1: decode anchors, per-anchor class sums ----------------
  for (int a = tid; a < NA; a += BTHREADS){
    const float* bp; int local, HW, W; float s;
    if (a < NL0){ bp = base0; local = a;       HW = NL0; W = 40; s = 16.f; }
    else        { bp = base1; local = a - NL0; HW = 400; W = 20; s = 32.f; }
    const int gx = local % W, gy = local / W;
    const float tx = bp[local], ty = bp[HW + local];
    const float tw = bp[2*HW + local], th = bp[3*HW + local];
    const float ob = bp[4*HW + local];
    sBX[a] = (tx + (float)gx) * s;
    sBY[a] = (ty + (float)gy) * s;
    sBW[a] = __expf(tw) * s;
    sBH[a] = __expf(th) * s;
    const float so = sigm(ob);
    sObj[a] = so;
    float ss = 0.f;
    #pragma unroll 4
    for (int c = 0; c < NC; ++c){
      const float p = sqrtf(sigm(bp[(5 + c)*HW + local]) * so);
      ss -= __logf(1.0f - p + 1e-12f);
    }
    sS[a]   = ss;
    sCnt[a] = 0;
    sSel[a] = 0;
    sBest[a] = ~0ull;
  }
  const int ng = ngt[b];
  for (int m = tid; m < NGTS; m += BTHREADS){
    const size_t o = ((size_t)b * NGTS + m) * 4;
    sGx[m] = gboxes[o + 0] * 640.f;
    sGy[m] = gboxes[o + 1] * 640.f;
    sGw[m] = gboxes[o + 2] * 640.f;
    sGh[m] = gboxes[o + 3] * 640.f;
    sGc[m] = gcls[b * NGTS + m];
    sGv[m] = (m < ng) ? 1 : 0;
  }
  __syncthreads();

  // ---------------- Phase 2: fg candidates (any gt: in_box | in_ctr) ----------------
  for (int a = tid; a < NA; a += BTHREADS){
    float axc, ayc, st; anchor_geom(a, axc, ayc, st);
    const float r = 2.5f * st;
    int any = 0;
    for (int m = 0; m < NGTS && !any; ++m){
      if (!sGv[m]) continue;
      const float cx = sGx[m], cy = sGy[m];
      const float hw = 0.5f * sGw[m], hh = 0.5f * sGh[m];
      const bool inb = (axc > cx - hw) && (axc < cx + hw) && (ayc > cy - hh) && (ayc < cy + hh);
      const bool inc = (axc > cx - r)  && (axc < cx + r)  && (ayc > cy - r)  && (ayc < cy + r);
      any = (int)(inb | inc);
    }
    sFg[a] = (unsigned char)any;
  }
  __syncthreads();

  // ---------------- Phase 3: SimOTA dynamic-k assignment, 1 wave per GT row ----------------
  for (int m = wave; m < NGTS; m += 8){
    if (!sGv[m]) continue;
    const float cx = sGx[m], cy = sGy[m], gw = sGw[m], gh = sGh[m];
    const int   cm = sGc[m];
    const float garea = gw * gh;
    const float glx = cx - 0.5f*gw, grx = cx + 0.5f*gw;
    const float gty = cy - 0.5f*gh, gby = cy + 0.5f*gh;

    // Prefetch this GT's class channel (read repeatedly during selection) into
    // near caches while the top-k IoU pass runs: one global_prefetch_b8 per 64B line.
    for (int a = lane * 16; a < NA; a += 32 * 16){
      const float* p = (a < NL0) ? &base0[(5 + cm)*NL0 + a] : &base1[(5 + cm)*400 + (a - NL0)];
      __builtin_prefetch(p, 0, 3);
    }

    // -- top-10 IoU sum -> dynamic k --
    sMask[wave][lane]      = 0u;
    sMask[wave][lane + 32] = 0u;
    float ksum = 0.f;
    for (int t = 0; t < 10; ++t){
      unsigned long long best = 0ull;  // pack (mono(iou)<<32)|a, max-reduce
      for (int a = lane; a < NA; a += 32){
        if ((sMask[wave][a >> 5] >> (a & 31)) & 1u) continue;
        float iou = 0.f;
        if (sFg[a]){
          const float ax = sBX[a], ay = sBY[a], aw = sBW[a], ah = sBH[a];
          const float tlx = fmaxf(ax - 0.5f*aw, glx), tly = fmaxf(ay - 0.5f*ah, gty);
          const float brx = fminf(ax + 0.5f*aw, grx), bry = fminf(ay + 0.5f*ah, gby);
          const float dx = brx - tlx, dy = bry - tly;
          const float inter = (dx > 0.f && dy > 0.f) ? dx * dy : 0.f;
          iou = inter / (aw*ah + garea - inter + 1e-16f);
        }
        const unsigned long long pk = ((unsigned long long)f2mono(iou) << 32) | (unsigned)a;
        if (pk > best) best = pk;
      }
      for (int off = 16; off; off >>= 1){
        const unsigned long long o = shflxor64(best, off);
        if (o > best) best = o;
      }
      const float v = mono2f((unsigned)(best >> 32));
      if (v <= 0.f) break;           // remaining candidates contribute 0 to the sum
      ksum += v;
      const int sa = (int)(best & 0xFFFFFFFFull);
      if (lane == 0) sMask[wave][sa >> 5] |= (1u << (sa & 31));
    }
    int dk = (int)ksum;              // trunc toward zero, matches .astype(int32)
    if (dk < 1)  dk = 1;
    if (dk > 10) dk = 10;

    // -- select dk minimum-cost anchors (stable ties -> smaller anchor index) --
    sMask[wave][lane]      = 0u;
    sMask[wave][lane + 32] = 0u;
    for (int t = 0; t < dk; ++t){
      unsigned long long best = ~0ull; // pack (mono(cost)<<32)|a, min-reduce
      for (int a = lane; a < NA; a += 32){
        if ((sMask[wave][a >> 5] >> (a & 31)) & 1u) continue;
        // pairwise IoU
        const float ax = sBX[a], ay = sBY[a], aw = sBW[a], ah = sBH[a];
        const float tlx = fmaxf(ax - 0.5f*aw, glx), tly = fmaxf(ay - 0.5f*ah, gty);
        const float brx = fminf(ax + 0.5f*aw, grx), bry = fminf(ay + 0.5f*ah, gby);
        const float dx = brx - tlx, dy = bry - tly;
        const float inter = (dx > 0.f && dy > 0.f) ? dx * dy : 0.f;
        const float iou = inter / (aw*ah + garea - inter + 1e-16f);
        // class cost: S[a] - log(p_c+eps) + log(1-p_c+eps), p_c from L2-resident channel read
        const float xl = (a < NL0) ? base0[(5 + cm)*NL0 + a] : base1[(5 + cm)*400 + (a - NL0)];
        const float p  = sqrtf(sigm(xl) * sObj[a]);
        float cost = sS[a] - __logf(p + 1e-12f) + __logf(1.f - p + 1e-12f)
                   - 3.0f * __logf(iou + 1e-8f);
        float axc, ayc, st; anchor_geom(a, axc, ayc, st);
        const float r = 2.5f * st;
        const bool inb = (axc > glx) && (axc < grx) && (ayc > gty) && (ayc < gby);
        const bool inc = (axc > cx - r) && (axc < cx + r) && (ayc > cy - r) && (ayc < cy + r);
        if (!(inb && inc)) cost += 100000.0f;
        if (!sFg[a])       cost += 1.0e9f;
        const unsigned km = f2mono(cost);
        if (t == 0)  // full-row pass: per-anchor argmin over GT rows (conflict resolution)
          atomicMin(&sBest[a], ((unsigned long long)km << 32) | (unsigned)m);
        const unsigned long long pk = ((unsigned long long)km << 32) | (unsigned)a;
        if (pk < best) best = pk;
      }
      for (int off = 16; off; off >>= 1){
        const unsigned long long o = shflxor64(best, off);
        if (o < best) best = o;
      }
      const int sa = (int)(best & 0xFFFFFFFFull);
      if (lane == 0){
        sMask[wave][sa >> 5] |= (1u << (sa & 31));
        atomicAdd(&sCnt[sa], 1);
        sSel[sa] = m;
      }
    }
  }
  __syncthreads();

  // ---------------- Phase 4: losses ----------------
  float liou = 0.f, lobj = 0.f, lcls = 0.f, lfg = 0.f;
  for (int a = tid; a < NA; a += BTHREADS){
    const int cnt = sCnt[a];
    const int fg  = (cnt > 0);
    const float ol = (a < NL0) ? base0[4*NL0 + a] : base1[4*400 + (a - NL0)];
    lobj += bce_logits(ol, fg ? 1.f : 0.f);
    if (fg){
      const int g = (cnt > 1) ? (int)(sBest[a] & 0xFFFFFFFFull) : sSel[a];
      const float ax = sBX[a], ay = sBY[a], aw = sBW[a], ah = sBH[a];
      const float gx = sGx[g], gy = sGy[g], gw = sGw[g], gh = sGh[g];
      const float tlx = fmaxf(ax - 0.5f*aw, gx - 0.5f*gw), tly = fmaxf(ay - 0.5f*ah, gy - 0.5f*gh);
      const float brx = fminf(ax + 0.5f*aw, gx + 0.5f*gw), bry = fminf(ay + 0.5f*ah, gy + 0.5f*gh);
      const float dx = brx - tlx, dy = bry - tly;
      const float inter = (dx > 0.f && dy > 0.f) ? dx * dy : 0.f;
      const float iou = inter / (aw*ah + gw*gh - inter + 1e-16f);
      liou += 1.0f - iou * iou;
      lfg  += 1.0f;
      const int cm = sGc[g];
      #pragma unroll 4
      for (int j = 0; j < NC; ++j){
        const float xl = (a < NL0) ? base0[(5 + j)*NL0 + a] : base1[(5 + j)*400 + (a - NL0)];
        lcls += bce_logits(xl, (j == cm) ? iou : 0.f);
      }
    }
  }
  sRed[0][tid] = liou;
  sRed[1][tid] = lobj;
  sRed[2][tid] = lcls;
  sRed[3][tid] = lfg;
  __syncthreads();

  // ---------------- Phase 5: exact f32 matrix-pipe block reduction (wave 0) ----------------
  // V_WMMA_F32_16X16X4_F32: A = ones(16x4) f32, B = 4x16 f32 (64 partials / WMMA),
  // C/D f32 accumulate. Chain 4 WMMAs to fold 256 partials into 16 column sums;
  // since all 16 columns are then summed, the exact B (k,n) lane mapping is
  // irrelevant to the final total — the columns partition the 64 values.
  if (tid < 32){
    v2f ones2;
    ones2.x = 1.0f; ones2.y = 1.0f;
    #pragma unroll
    for (int q = 0; q < 4; ++q){
      v8f c = {};
      #pragma unroll
      for (int i = 0; i < 4; ++i){
        v2f bm;
        bm.x = sRed[q][i*64 + lane];
        bm.y = sRed[q][i*64 + 32 + lane];
        c = __builtin_amdgcn_wmma_f32_16x16x4_f32(false, ones2, false, bm,
                                                  (short)0, c, false, false);
      }
      float v = c[0];                         // row M=0: column sums n=0..15 in lanes 0..15
      for (int off = 8; off; off >>= 1) v += __shfl_xor(v, off, 32);
      if (lane == 0) atomicAdd(&acc[q], v);
    }
  }
}

__global__ void init_acc_kernel(float* acc){
  if (threadIdx.x < 8) acc[threadIdx.x] = 0.f;
}

__global__ void finalize_kernel(const float* __restrict__ acc,
                                const int* __restrict__ ngt,
                                float* __restrict__ out){
  if (threadIdx.x == 0){
    const float siou = acc[0], sobj = acc[1], scls = acc[2], sfg = acc[3];
    const float nfg = fmaxf(sfg, 1.0f);
    const float liou5 = 5.0f * siou / nfg;
    const float lobj  = sobj / nfg;
    const float lcls  = scls / nfg;
    int tg = 0;
    for (int i = 0; i < 64; ++i) tg += ngt[i];
    out[0] = liou5 + lobj + lcls;
    out[1] = liou5;
    out[2] = lobj;
    out[3] = lcls;
    out[4] = nfg / fmaxf((float)tg, 1.0f);
  }
}

extern "C" void kernel_launch(void* const* d_in, const int* in_sizes, int n_in,
                              void* d_out, int out_size, void* d_ws, size_t ws_size,
                              hipStream_t stream) {
  const float* l0     = (const float*)d_in[0];   // (64,25,40,40)
  const float* l1     = (const float*)d_in[1];   // (64,25,20,20)
  const float* gboxes = (const float*)d_in[2];   // (64,100,4)
  const int*   gcls   = (const int*)  d_in[3];   // (64,100)
  const int*   ngt    = (const int*)  d_in[4];   // (64,)
  float* acc = (float*)d_ws;                     // [iou, obj, cls, fg]
  float* out = (float*)d_out;                    // 5 scalars

  init_acc_kernel<<<1, 64, 0, stream>>>(acc);
  yolox_loss_kernel<<<64, BTHREADS, 0, stream>>>(l0, l1, gboxes, gcls, ngt, acc);
  finalize_kernel<<<1, 64, 0, stream>>>(acc, ngt, out);
}